// Semantic_Mapping_78683800862860
// MI455X (gfx1250) — compile-verified
//
#include <hip/hip_runtime.h>
#include <math.h>

typedef __attribute__((ext_vector_type(2))) float v2f;
typedef __attribute__((ext_vector_type(8))) float v8f;

#define H_IMG 480
#define W_IMG 640
#define HW_IMG (H_IMG * W_IMG)
#define VR_ 100
#define ZDIM_ 80
#define NF_ 6
#define FSTRIDE (VR_ * VR_ * ZDIM_)   /* 800000 floats per feature */
#define MSZ_ 960
#define M2_ (MSZ_ * MSZ_)
#define CH_ 9

/* workspace layout (float offsets) */
#define GRID_SZ (NF_ * FSTRIDE)                 /* 4,800,000 */
#define AV_OFF  GRID_SZ
#define ROT_OFF (AV_OFF + CH_ * M2_)
#define PRM_OFF (ROT_OFF + CH_ * M2_)
#define WS_TOTAL (PRM_OFF + 16)
/* prm: [0]=ct [1]=sn [2]=sx [3]=sy [4..7]=ks (bits, nonneg floats) */

/* output layout (float offsets): fp_map | map_pred | pose | pose */
#define OUT_MAP   10000
#define OUT_POSE1 (OUT_MAP + CH_ * M2_)
#define OUT_POSE2 (OUT_POSE1 + 3)

__global__ void zero_kernel(float* __restrict__ p, int n) {
    int i = blockIdx.x * 256 + threadIdx.x;
    if (i < n) p[i] = 0.0f;
}

__global__ void pose_kernel(const float* __restrict__ pose_obs,
                            const float* __restrict__ poses_last,
                            float* __restrict__ prm, float* __restrict__ out) {
    if (threadIdx.x != 0) return;
    const float DEGc = 57.29577951308232f;
    float th = poses_last[2] / DEGc;
    float s = sinf(th), c = cosf(th);
    float ny = poses_last[1] + pose_obs[0] * s + pose_obs[1] * c;
    float nx = poses_last[0] + pose_obs[0] * c - pose_obs[1] * s;
    float nt = poses_last[2] + pose_obs[2] * DEGc;
    nt = fmodf(nt - 180.0f, 360.0f) + 180.0f;
    nt = fmodf(nt + 180.0f, 360.0f) - 180.0f;
    out[OUT_POSE1 + 0] = nx; out[OUT_POSE1 + 1] = ny; out[OUT_POSE1 + 2] = nt;
    out[OUT_POSE2 + 0] = nx; out[OUT_POSE2 + 1] = ny; out[OUT_POSE2 + 2] = nt;
    const float half = 480.0f;
    float sx = -(nx * 100.0f / 5.0f - half) / half;
    float sy = -(ny * 100.0f / 5.0f - half) / half;
    float sth = (90.0f - nt) * 0.017453292519943295f;
    prm[0] = cosf(sth); prm[1] = sinf(sth); prm[2] = sx; prm[3] = sy;
    unsigned* kb = (unsigned*)(prm + 4);
    kb[0] = 0u; kb[1] = 0u; kb[2] = 0u; kb[3] = 0u;
}

/* ks[q] = max over all pixels of obs channel 4+q (q = 0..3); values >= 0 so
   uint-bit atomicMax == float max. */
__global__ void ks_kernel(const float* __restrict__ obs, unsigned* __restrict__ ksbits) {
    __shared__ float sm[4][256];
    int t = threadIdx.x;
    float m0 = 0.f, m1 = 0.f, m2 = 0.f, m3 = 0.f;
    for (int i = blockIdx.x * 256 + t; i < HW_IMG; i += gridDim.x * 256) {
        m0 = fmaxf(m0, obs[4 * HW_IMG + i]);
        m1 = fmaxf(m1, obs[5 * HW_IMG + i]);
        m2 = fmaxf(m2, obs[6 * HW_IMG + i]);
        m3 = fmaxf(m3, obs[7 * HW_IMG + i]);
    }
    sm[0][t] = m0; sm[1][t] = m1; sm[2][t] = m2; sm[3][t] = m3;
    __syncthreads();
    for (int s = 128; s > 0; s >>= 1) {
        if (t < s)
            for (int q = 0; q < 4; ++q) sm[q][t] = fmaxf(sm[q][t], sm[q][t + s]);
        __syncthreads();
    }
    if (t == 0)
        for (int q = 0; q < 4; ++q) atomicMax(&ksbits[q], __float_as_uint(sm[q][0]));
}

/* trilinear scatter-add of [1, obs[4..8]] into grid[f][x][y][z] */
__global__ void splat_kernel(const float* __restrict__ obs, float* __restrict__ grid) {
    int idx = blockIdx.x * 256 + threadIdx.x;
    if (idx >= HW_IMG) return;
    int i = idx / W_IMG, j = idx % W_IMG;
    float depth = obs[3 * HW_IMG + idx];
    const float fpix = 320.0f / tanf(39.5f * 0.017453292519943295f);
    float X = ((float)j - 319.5f) * depth / fpix + 250.0f;
    float Z = ((float)(479 - i) - 239.5f) * depth / fpix + 88.0f;
    float px = X / 5.0f;              /* pos in dim0 (x) */
    float py = depth / 5.0f;          /* pos in dim1 (y) */
    float pz = Z / 5.0f + 8.0f;       /* pos in dim2 (z) */
    float fx = floorf(px), fy = floorf(py), fz = floorf(pz);
    float f1 = obs[4 * HW_IMG + idx], f2 = obs[5 * HW_IMG + idx],
          f3 = obs[6 * HW_IMG + idx], f4 = obs[7 * HW_IMG + idx],
          f5 = obs[8 * HW_IMG + idx];
    #pragma unroll
    for (int c0 = 0; c0 < 2; ++c0)
    #pragma unroll
    for (int c1 = 0; c1 < 2; ++c1)
    #pragma unroll
    for (int c2 = 0; c2 < 2; ++c2) {
        float qx = fx + c0, qy = fy + c1, qz = fz + c2;
        if (!(qx > 0.f && qx < 100.f && qy > 0.f && qy < 100.f && qz > 0.f && qz < 80.f))
            continue;  /* reference weight would be exactly 0 */
        float w = (1.f - fabsf(px - qx)) * (1.f - fabsf(py - qy)) * (1.f - fabsf(pz - qz));
        int gi = (((int)qx) * VR_ + (int)qy) * ZDIM_ + (int)qz;
        atomicAdd(&grid[gi], w);
        atomicAdd(&grid[1 * FSTRIDE + gi], w * f1);
        atomicAdd(&grid[2 * FSTRIDE + gi], w * f2);
        atomicAdd(&grid[3 * FSTRIDE + gi], w * f3);
        atomicAdd(&grid[4 * FSTRIDE + gi], w * f4);
        atomicAdd(&grid[5 * FSTRIDE + gi], w * f5);
    }
}

/* z-reduction via V_WMMA_F32_16X16X4_F32.
   Wave handles 16 cells (flat cell = x*100+y) of feature f.
   A (16x4): lane L<16 -> A[L][0..1], lane L+16 -> A[L][2..3].
   B (4x16): col 0 = ones (allp), col 1 = z in [13,25) mask (ahp).
   D: column n lives in lane n (rows 0-7) / lane n+16 (rows 8-15). */
__global__ void proj_kernel(const float* __restrict__ grid, float* __restrict__ av,
                            float* __restrict__ out) {
    int wave = blockIdx.x;                 /* blockDim = 32 */
    int f = wave / 625;
    int cellbase = (wave % 625) * 16;
    int lane = threadIdx.x;
    int m = lane & 15;
    int hi = lane >> 4;
    const float* base = grid + ((size_t)(f * 10000 + cellbase + m)) * ZDIM_;
    v8f acc = {};
    for (int k = 0; k < 20; ++k) {
        int z0 = 4 * k + 2 * hi;
        v2f a, b;
        a.x = rintf(base[z0]);             /* jnp.round == RTE */
        a.y = rintf(base[z0 + 1]);
        float mk0 = (z0 >= 13 && z0 < 25) ? 1.0f : 0.0f;
        float mk1 = (z0 + 1 >= 13 && z0 + 1 < 25) ? 1.0f : 0.0f;
        b.x = (m == 0) ? 1.0f : ((m == 1) ? mk0 : 0.0f);
        b.y = (m == 0) ? 1.0f : ((m == 1) ? mk1 : 0.0f);
        acc = __builtin_amdgcn_wmma_f32_16x16x4_f32(false, a, false, b,
                                                    (short)0, acc, false, false);
    }
    if (m <= 1) {
        #pragma unroll
        for (int r = 0; r < 8; ++r) {
            int cell = cellbase + hi * 8 + r;
            int x = cell / 100, y = cell % 100;
            size_t pos = (size_t)(480 + y) * MSZ_ + (430 + x);
            float v = acc[r];
            if (m == 1) {  /* ahp */
                if (f == 0) {
                    float fp = fminf(fmaxf(v, 0.f), 1.f);
                    out[y * 100 + x] = fp;                       /* fp_map */
                    av[pos] = fp;                                /* channel 0 */
                } else {
                    av[(size_t)(3 + f) * M2_ + pos] = fminf(fmaxf(v / 5.0f, 0.f), 1.f);
                }
            } else {       /* allp */
                if (f == 0)
                    av[(size_t)1 * M2_ + pos] = fminf(fmaxf(v, 0.f), 1.f);
            }
        }
    }
}

__device__ __forceinline__ void bilin_setup(float u, float v, int* off, float* wt) {
    float x = (u + 1.0f) * 0.5f * 959.0f;
    float y = (v + 1.0f) * 0.5f * 959.0f;
    float x0 = floorf(x), y0 = floorf(y);
    int k = 0;
    #pragma unroll
    for (int dy = 0; dy < 2; ++dy)
    #pragma unroll
    for (int dx = 0; dx < 2; ++dx) {
        float xx = x0 + dx, yy = y0 + dy;
        float w = (1.0f - fabsf(x - xx)) * (1.0f - fabsf(y - yy));
        bool inb = (xx >= 0.f) && (xx <= 959.f) && (yy >= 0.f) && (yy <= 959.f);
        int cxi = (int)fminf(fmaxf(xx, 0.f), 959.f);
        int cyi = (int)fminf(fmaxf(yy, 0.f), 959.f);
        wt[k] = inb ? w : 0.0f;
        off[k] = cyi * MSZ_ + cxi;
        ++k;
    }
}

__global__ void rotate_kernel(const float* __restrict__ av, float* __restrict__ rot,
                              const float* __restrict__ prm) {
    int idx = blockIdx.x * 256 + threadIdx.x;
    if (idx >= M2_) return;
    int i = idx / MSZ_, j = idx % MSZ_;
    float ct = prm[0], sn = prm[1];
    float gx = 2.0f * (float)j / 959.0f - 1.0f;
    float gy = 2.0f * (float)i / 959.0f - 1.0f;
    float u = ct * gx - sn * gy;
    float v = sn * gx + ct * gy;
    int off[4]; float wt[4];
    bilin_setup(u, v, off, wt);
    for (int c = 0; c < CH_; ++c) {
        const float* img = av + (size_t)c * M2_;
        float s = img[off[0]] * wt[0] + img[off[1]] * wt[1] +
                  img[off[2]] * wt[2] + img[off[3]] * wt[3];
        rot[(size_t)c * M2_ + idx] = s;
    }
}

__global__ void translate_kernel(const float* __restrict__ rot,
                                 const float* __restrict__ maps_last,
                                 const float* __restrict__ prm,
                                 float* __restrict__ out) {
    int idx = blockIdx.x * 256 + threadIdx.x;
    if (idx >= M2_) return;
    int i = idx / MSZ_, j = idx % MSZ_;
    float gx = 2.0f * (float)j / 959.0f - 1.0f;
    float gy = 2.0f * (float)i / 959.0f - 1.0f;
    float u = gx + prm[2];
    float v = gy + prm[3];
    int off[4]; float wt[4];
    bilin_setup(u, v, off, wt);
    for (int c = 0; c < CH_; ++c) {
        const float* img = rot + (size_t)c * M2_;
        float t = img[off[0]] * wt[0] + img[off[1]] * wt[1] +
                  img[off[2]] * wt[2] + img[off[3]] * wt[3];
        if (c >= 4 && c < 8) {
            float ksv = prm[4 + (c - 4)];
            t = (t > 0.0f) ? ksv : t;
        }
        out[OUT_MAP + (size_t)c * M2_ + idx] = fmaxf(maps_last[(size_t)c * M2_ + idx], t);
    }
}

extern "C" void kernel_launch(void* const* d_in, const int* in_sizes, int n_in,
                              void* d_out, int out_size, void* d_ws, size_t ws_size,
                              hipStream_t stream) {
    const float* obs        = (const float*)d_in[0];
    const float* pose_obs   = (const float*)d_in[1];
    const float* maps_last  = (const float*)d_in[2];
    const float* poses_last = (const float*)d_in[3];
    float* out = (float*)d_out;
    float* ws  = (float*)d_ws;
    if (ws_size < (size_t)WS_TOTAL * sizeof(float)) return;

    float* grid = ws;
    float* av   = ws + AV_OFF;
    float* rot  = ws + ROT_OFF;
    float* prm  = ws + PRM_OFF;

    int nzero = GRID_SZ + CH_ * M2_;   /* grid + agent_view (contiguous) */
    zero_kernel<<<(nzero + 255) / 256, 256, 0, stream>>>(ws, nzero);
    pose_kernel<<<1, 32, 0, stream>>>(pose_obs, poses_last, prm, out);
    ks_kernel<<<512, 256, 0, stream>>>(obs, (unsigned*)(prm + 4));
    splat_kernel<<<(HW_IMG + 255) / 256, 256, 0, stream>>>(obs, grid);
    proj_kernel<<<NF_ * 625, 32, 0, stream>>>(grid, av, out);
    rotate_kernel<<<(M2_ + 255) / 256, 256, 0, stream>>>(av, rot, prm);
    translate_kernel<<<(M2_ + 255) / 256, 256, 0, stream>>>(rot, maps_last, prm, out);
}